// RelationshipAttention_23149873726098
// MI455X (gfx1250) — compile-verified
//
#include <hip/hip_runtime.h>
#include <hip/hip_bf16.h>
#include <cmath>

typedef float v2f __attribute__((ext_vector_type(2)));
typedef float v8f __attribute__((ext_vector_type(8)));

#define BATCH 16
#define NTOK  2048
#define DIM   1024
#define KINST 100
#define RREL  5
#define KTILE 7            // ceil(100/16) padded tiles
#define MASKV 1000000000.0f
#define JGRP  8            // column tiles processed concurrently per wave

// ---------------------------------------------------------------------------
// Kernel 1: online-softmax row statistics of scores = Q K^T (per batch).
// One wave owns a 16-row strip. Register-blocked: 8 column-tile f32
// accumulators live at once; each K-step loads 1 A-frag + 8 B-frags and
// issues 8 back-to-back V_WMMA_F32_16X16X4_F32 (1.125 loads/WMMA, partial
// loadcnt waits instead of wait-0 per WMMA). Next k-group prefetched.
// A-frag (16x4 f32): lane L holds row M=L&15; VGPR v holds K = v + 2*(L>>4).
// C/D: VGPR r -> row r (lanes 0-15) / r+8 (lanes 16-31), N = lane&15.
// ---------------------------------------------------------------------------
__global__ __launch_bounds__(256) void qk_row_stats(
    const float* __restrict__ q, const float* __restrict__ k,
    float* __restrict__ m_buf, float* __restrict__ l_buf,
    float* __restrict__ dsm_buf)
{
    const int wave  = threadIdx.x >> 5;
    const int lane  = threadIdx.x & 31;
    const int strip_global = blockIdx.x * 8 + wave;     // B * 128 strips total
    const int b     = strip_global >> 7;
    const int strip = strip_global & 127;

    const int row15 = lane & 15;
    const int half  = lane >> 4;

    // per-lane base pointers (fragment gather pattern baked in)
    const float* __restrict__ qlane =
        q + ((size_t)b * NTOK + strip * 16 + row15) * DIM + 2 * half;
    const float* __restrict__ klane =
        k + ((size_t)b * NTOK + row15) * DIM + 2 * half;

    float m[8], l[8], diag_raw[8];
    #pragma unroll
    for (int r = 0; r < 8; ++r) { m[r] = -INFINITY; l[r] = 0.f; diag_raw[r] = 0.f; }

    for (int jtg = 0; jtg < 128 / JGRP; ++jtg) {
        const float* __restrict__ kg = klane + (size_t)jtg * JGRP * 16 * DIM;
        if (jtg < 128 / JGRP - 1)
            __builtin_prefetch(kg + (size_t)JGRP * 16 * DIM, 0, 1);

        v8f acc[JGRP] = {};
        for (int kk = 0; kk < DIM; kk += 4) {
            v2f a = *(const v2f*)(qlane + kk);
            v2f bf[JGRP];
            #pragma unroll
            for (int t = 0; t < JGRP; ++t)
                bf[t] = *(const v2f*)(kg + (size_t)t * 16 * DIM + kk);
            #pragma unroll
            for (int t = 0; t < JGRP; ++t)
                acc[t] = __builtin_amdgcn_wmma_f32_16x16x4_f32(
                    false, a, false, bf[t], (short)0, acc[t], false, false);
        }

        // capture raw diagonal dot (tile index == strip); uniform branch
        #pragma unroll
        for (int t = 0; t < JGRP; ++t) {
            if (jtg * JGRP + t == strip) {
                #pragma unroll
                for (int r = 0; r < 8; ++r)
                    diag_raw[r] = __shfl((float)acc[t][r], r + 8 * half, 16);
            }
        }

        // online softmax update per tile, per row (16-lane half reductions)
        #pragma unroll
        for (int t = 0; t < JGRP; ++t) {
            #pragma unroll
            for (int r = 0; r < 8; ++r) {
                float v = acc[t][r];
                float tmax = v;
                for (int s = 8; s >= 1; s >>= 1)
                    tmax = fmaxf(tmax, __shfl_xor(tmax, s, 16));
                float mn = fmaxf(m[r], tmax);
                float tsum = __expf(v - mn);
                for (int s = 8; s >= 1; s >>= 1)
                    tsum += __shfl_xor(tsum, s, 16);
                l[r] = l[r] * __expf(m[r] - mn) + tsum;
                m[r] = mn;
            }
        }
    }

    if (row15 == 0) {       // lane 0 writes rows 0..7, lane 16 writes rows 8..15
        const int row_base = strip * 16 + 8 * half;
        #pragma unroll
        for (int r = 0; r < 8; ++r) {
            const int row = row_base + r;
            m_buf[b * NTOK + row]   = m[r];
            l_buf[b * NTOK + row]   = l[r];
            dsm_buf[b * NTOK + row] = __expf(diag_raw[r] - m[r]) / l[r];
        }
    }
}

// ---------------------------------------------------------------------------
// Kernel 2: per-batch top-100 of diag softmax, indices sorted ascending.
// Iterative argmax with first-occurrence tie-break (matches lax.top_k).
// ---------------------------------------------------------------------------
__global__ __launch_bounds__(256) void topk_inst(
    const float* __restrict__ dsm_buf, int* __restrict__ inst)
{
    __shared__ float sval[NTOK];
    __shared__ float rv[256];
    __shared__ int   ri[256];
    __shared__ int   chosen[KINST];
    const int b = blockIdx.x, t = threadIdx.x;

    for (int i = t; i < NTOK; i += 256) sval[i] = dsm_buf[b * NTOK + i];
    __syncthreads();

    for (int kk = 0; kk < KINST; ++kk) {
        float bv = -INFINITY; int bi = NTOK;
        for (int i = t; i < NTOK; i += 256) {
            float v = sval[i];
            if (v > bv) { bv = v; bi = i; }
        }
        rv[t] = bv; ri[t] = bi;
        __syncthreads();
        for (int s = 128; s > 0; s >>= 1) {
            if (t < s) {
                float ov = rv[t + s]; int oi = ri[t + s];
                if (ov > rv[t] || (ov == rv[t] && oi < ri[t])) { rv[t] = ov; ri[t] = oi; }
            }
            __syncthreads();
        }
        if (t == 0) { chosen[kk] = ri[0]; sval[ri[0]] = -INFINITY; }
        __syncthreads();
    }

    if (t == 0) {
        for (int i = 1; i < KINST; ++i) {            // insertion sort ascending
            int key = chosen[i]; int j = i - 1;
            while (j >= 0 && chosen[j] > key) { chosen[j + 1] = chosen[j]; --j; }
            chosen[j + 1] = key;
        }
        for (int i = 0; i < KINST; ++i) inst[b * KINST + i] = chosen[i];
    }
}

// ---------------------------------------------------------------------------
// Kernel 3: rel[i][j] = exp(dot(q_Ii, k_Ij) - m_Ii) / l_Ii  (diag -> 1e9),
// recomputed via WMMA over 7x7 padded 16x16 tiles. One wave per tile.
// ---------------------------------------------------------------------------
__global__ __launch_bounds__(32) void rel_wmma(
    const float* __restrict__ q, const float* __restrict__ k,
    const float* __restrict__ m_buf, const float* __restrict__ l_buf,
    const int* __restrict__ inst, float* __restrict__ rel)
{
    const int tile = blockIdx.x % (KTILE * KTILE);
    const int b    = blockIdx.x / (KTILE * KTILE);
    const int ti = tile / KTILE, tj = tile % KTILE;
    const int lane = threadIdx.x & 31;
    const int row15 = lane & 15, half = lane >> 4;

    const int iIdx = ti * 16 + row15;
    const int jIdx = tj * 16 + row15;
    const int tokI = inst[b * KINST + (iIdx < KINST - 1 ? iIdx : KINST - 1)];
    const int tokJ = inst[b * KINST + (jIdx < KINST - 1 ? jIdx : KINST - 1)];

    const float* __restrict__ qrow = q + ((size_t)b * NTOK + tokI) * DIM + 2 * half;
    const float* __restrict__ krow = k + ((size_t)b * NTOK + tokJ) * DIM + 2 * half;

    v8f acc = {};
    for (int kk = 0; kk < DIM; kk += 16) {
        v2f a[4], bf[4];
        #pragma unroll
        for (int u = 0; u < 4; ++u) {
            a[u]  = *(const v2f*)(qrow + kk + 4 * u);
            bf[u] = *(const v2f*)(krow + kk + 4 * u);
        }
        #pragma unroll
        for (int u = 0; u < 4; ++u)
            acc = __builtin_amdgcn_wmma_f32_16x16x4_f32(
                false, a[u], false, bf[u], (short)0, acc, false, false);
    }

    const int J = tj * 16 + row15;
    #pragma unroll
    for (int r = 0; r < 8; ++r) {
        const int I = ti * 16 + r + 8 * half;
        if (I < KINST && J < KINST) {
            float val;
            if (I == J) {
                val = MASKV;
            } else {
                const int tok = inst[b * KINST + I];
                const float mi = m_buf[b * NTOK + tok];
                const float li = l_buf[b * NTOK + tok];
                val = __expf((float)acc[r] - mi) / li;
            }
            rel[((size_t)b * KINST + I) * KINST + J] = val;
        }
    }
}

// ---------------------------------------------------------------------------
// Kernel 4: top-5 per rel row (first-occurrence tie-break), cols sorted
// ascending; emit (b, subj, obj) triples as floats + int pairs to scratch.
// ---------------------------------------------------------------------------
__global__ __launch_bounds__(256) void toprel(
    const float* __restrict__ rel, const int* __restrict__ inst,
    int* __restrict__ pairs, float* __restrict__ out_idx)
{
    const int tid = blockIdx.x * blockDim.x + threadIdx.x;
    if (tid >= BATCH * KINST) return;
    const int b = tid / KINST, i = tid % KINST;
    const float* __restrict__ row = rel + ((size_t)b * KINST + i) * KINST;

    int sel[RREL];
    for (int r = 0; r < RREL; ++r) {
        float bv = -INFINITY; int bi = 0;
        for (int j = 0; j < KINST; ++j) {
            bool skip = false;
            for (int s = 0; s < r; ++s) if (sel[s] == j) skip = true;
            if (skip) continue;
            float v = row[j];
            if (v > bv) { bv = v; bi = j; }
        }
        sel[r] = bi;
    }
    for (int a = 0; a < RREL; ++a)                     // sort cols ascending
        for (int c = a + 1; c < RREL; ++c)
            if (sel[c] < sel[a]) { int t = sel[a]; sel[a] = sel[c]; sel[c] = t; }

    const int subj = inst[b * KINST + i];
    for (int r = 0; r < RREL; ++r) {
        const int obj = inst[b * KINST + sel[r]];
        const int p   = i * RREL + r;
        const size_t base = ((size_t)(b * KINST * RREL + p)) * 3;
        out_idx[base + 0] = (float)b;
        out_idx[base + 1] = (float)subj;
        out_idx[base + 2] = (float)obj;
        pairs[(b * KINST * RREL + p) * 2 + 0] = subj;
        pairs[(b * KINST * RREL + p) * 2 + 1] = obj;
    }
}

// ---------------------------------------------------------------------------
// Kernel 5: layernorm(q[subj] + q[obj]) over d=1024. One block per pair.
// ---------------------------------------------------------------------------
__global__ __launch_bounds__(256) void ln_embed(
    const float* __restrict__ q, const int* __restrict__ pairs,
    float* __restrict__ out_emb)
{
    const int p = blockIdx.x;                 // 0 .. BATCH*500-1
    const int b = p / (KINST * RREL);
    const int t = threadIdx.x;
    const int subj = pairs[p * 2], obj = pairs[p * 2 + 1];
    const float* __restrict__ qs = q + ((size_t)b * NTOK + subj) * DIM;
    const float* __restrict__ qo = q + ((size_t)b * NTOK + obj) * DIM;

    float x[4];
    float s = 0.f, s2 = 0.f;
    #pragma unroll
    for (int u = 0; u < 4; ++u) {
        float v = qs[t + 256 * u] + qo[t + 256 * u];
        x[u] = v; s += v; s2 += v * v;
    }
    for (int off = 16; off >= 1; off >>= 1) {
        s  += __shfl_xor(s,  off, 32);
        s2 += __shfl_xor(s2, off, 32);
    }
    __shared__ float ss[8], ssq[8];
    const int wv = t >> 5, lane = t & 31;
    if (lane == 0) { ss[wv] = s; ssq[wv] = s2; }
    __syncthreads();
    if (t == 0) {
        float a = 0.f, a2 = 0.f;
        for (int w = 0; w < 8; ++w) { a += ss[w]; a2 += ssq[w]; }
        ss[0] = a; ssq[0] = a2;
    }
    __syncthreads();
    const float mean = ss[0] * (1.0f / DIM);
    const float var  = ssq[0] * (1.0f / DIM) - mean * mean;
    const float rstd = rsqrtf(var + 1e-5f);
    #pragma unroll
    for (int u = 0; u < 4; ++u)
        out_emb[(size_t)p * DIM + t + 256 * u] = (x[u] - mean) * rstd;
}

// ---------------------------------------------------------------------------
extern "C" void kernel_launch(void* const* d_in, const int* in_sizes, int n_in,
                              void* d_out, int out_size, void* d_ws, size_t ws_size,
                              hipStream_t stream)
{
    const float* q = (const float*)d_in[0];
    const float* k = (const float*)d_in[1];
    // d_in[2] = top_k_instances (100), d_in[3] = top_k_relationships (5): fixed.

    // workspace carve-out (~1 MB)
    float* m_buf = (float*)d_ws;
    float* l_buf = m_buf + BATCH * NTOK;
    float* dsm   = l_buf + BATCH * NTOK;
    int*   inst  = (int*)(dsm + BATCH * NTOK);
    float* rel   = (float*)(inst + BATCH * KINST);
    int*   pairs = (int*)(rel + BATCH * KINST * KINST);

    float* out     = (float*)d_out;
    float* out_emb = out + BATCH * KINST * RREL * 3;   // 24000 floats of indices

    qk_row_stats<<<BATCH * 128 / 8, 256, 0, stream>>>(q, k, m_buf, l_buf, dsm);
    topk_inst<<<BATCH, 256, 0, stream>>>(dsm, inst);
    rel_wmma<<<BATCH * KTILE * KTILE, 32, 0, stream>>>(q, k, m_buf, l_buf, inst, rel);
    toprel<<<(BATCH * KINST + 255) / 256, 256, 0, stream>>>(rel, inst, pairs, out);
    ln_embed<<<BATCH * KINST * RREL, 256, 0, stream>>>(q, pairs, out_emb);
}